// ATSS_Criteria_72524817760272
// MI455X (gfx1250) — compile-verified
//
#include <hip/hip_runtime.h>
#include <hip/hip_bf16.h>
#include <math.h>

// ---------------- problem constants (from reference) ----------------
#define A_TOTAL 21824
#define BATCH   16
#define MGT     32
#define NUM_FG  80
#define TOPK_K  9
#define NLEV    5
#define EPSF    1e-6f
#define THREADS 256
#define NB3     2048      // blocks for focal kernel (multiple of 64)
#define JS_W    0.1f

__constant__ int c_lev_start[NLEV] = {0, 16384, 20480, 21504, 21760};
__constant__ int c_lev_size [NLEV] = {16384, 4096, 1024, 256, 64};

typedef float vf4 __attribute__((ext_vector_type(4)));
typedef float v2f __attribute__((ext_vector_type(2)));
typedef float v8f __attribute__((ext_vector_type(8)));

// ---------------- helpers ----------------
__device__ __forceinline__ float fclampf(float x, float lo, float hi) {
    return fminf(fmaxf(x, lo), hi);
}

__device__ __forceinline__ float blockReduceSumF(float v, float* smem) {
    int lane = threadIdx.x & 31, wid = threadIdx.x >> 5;
#pragma unroll
    for (int o = 16; o > 0; o >>= 1) v += __shfl_down(v, o, 32);
    if (lane == 0) smem[wid] = v;
    __syncthreads();
    float r = (threadIdx.x < 8) ? smem[threadIdx.x] : 0.f;
    if (wid == 0) {
#pragma unroll
        for (int o = 4; o > 0; o >>= 1) r += __shfl_down(r, o, 32);
    }
    __syncthreads();
    return r;   // valid on thread 0
}

__device__ __forceinline__ int blockReduceSumI(int v, int* smem) {
    int lane = threadIdx.x & 31, wid = threadIdx.x >> 5;
#pragma unroll
    for (int o = 16; o > 0; o >>= 1) v += __shfl_down(v, o, 32);
    if (lane == 0) smem[wid] = v;
    __syncthreads();
    int r = (threadIdx.x < 8) ? smem[threadIdx.x] : 0;
    if (wid == 0) {
#pragma unroll
        for (int o = 4; o > 0; o >>= 1) r += __shfl_down(r, o, 32);
    }
    __syncthreads();
    return r;
}

__device__ __forceinline__ unsigned long long
blockMinU64(unsigned long long v, unsigned long long* smem) {
    int lane = threadIdx.x & 31, wid = threadIdx.x >> 5;
#pragma unroll
    for (int o = 16; o > 0; o >>= 1) {
        unsigned long long w = __shfl_down(v, o, 32);
        if (w < v) v = w;
    }
    if (lane == 0) smem[wid] = v;
    __syncthreads();
    unsigned long long r = (threadIdx.x < 8) ? smem[threadIdx.x] : ~0ULL;
    if (wid == 0) {
#pragma unroll
        for (int o = 4; o > 0; o >>= 1) {
            unsigned long long w = __shfl_down(r, o, 32);
            if (w < r) r = w;
        }
    }
    __syncthreads();
    return r;   // valid on thread 0
}

__device__ __forceinline__ void decode_box(float dx, float dy, float dw, float dh,
                                           float acx, float acy, float aw, float ah,
                                           float* o) {
    float cx = acx + dx * aw;
    float cy = acy + dy * ah;
    float w  = expf(fclampf(dw, -4.f, 4.f)) * aw;
    float h  = expf(fclampf(dh, -4.f, 4.f)) * ah;
    o[0] = cx - 0.5f * w; o[1] = cy - 0.5f * h;
    o[2] = cx + 0.5f * w; o[3] = cy + 0.5f * h;
}

__device__ __forceinline__ float giou_fn(const float* b1, const float* b2) {
    float tlx = fmaxf(b1[0], b2[0]), tly = fmaxf(b1[1], b2[1]);
    float brx = fminf(b1[2], b2[2]), bry = fminf(b1[3], b2[3]);
    float iw = fmaxf(brx - tlx, 0.f), ih = fmaxf(bry - tly, 0.f);
    float inter = iw * ih;
    float a1 = (b1[2] - b1[0]) * (b1[3] - b1[1]);
    float a2 = (b2[2] - b2[0]) * (b2[3] - b2[1]);
    float uni = fmaxf(a1 + a2 - inter, EPSF);
    float iou = inter / uni;
    float ex1 = fminf(b1[0], b2[0]), ey1 = fminf(b1[1], b2[1]);
    float ex2 = fmaxf(b1[2], b2[2]), ey2 = fmaxf(b1[3], b2[3]);
    float ew = fmaxf(ex2 - ex1, 0.f), eh = fmaxf(ey2 - ey1, 0.f);
    float enc = fmaxf(ew * eh, EPSF);
    return iou - (enc - uni) / enc;
}

// ---------------- kernel 1: ATSS assignment, one block per (b, gt) ----------------
__global__ void assign_kernel(const float* __restrict__ anchors,
                              const float* __restrict__ gt,       // B*M*5
                              unsigned long long* __restrict__ keys) {
    __shared__ unsigned long long s_red[8];
    __shared__ int   s_cand[NLEV * TOPK_K];
    __shared__ float s_iou [NLEV * TOPK_K];
    __shared__ float s_thr;

    int blk = blockIdx.x;
    int b = blk / MGT, g = blk % MGT;
    const float* G = gt + ((size_t)b * MGT + g) * 5;
    float gx1 = G[0], gy1 = G[1], gx2 = G[2], gy2 = G[3], glab = G[4];
    if (glab <= 0.f) return;              // uniform per block: invalid gt contributes nothing
    float gcx = (gx1 + gx2) * 0.5f, gcy = (gy1 + gy2) * 0.5f;
    int t = threadIdx.x;

    // per-level top-9 smallest center distances (9 rounds of block argmin,
    // packed (d^2, index) keys -> tie break by lowest index, matching top_k)
    for (int L = 0; L < NLEV; ++L) {
        int st = c_lev_start[L], n = c_lev_size[L];
        for (int r = 0; r < TOPK_K; ++r) {
            unsigned long long best = ~0ULL;
            for (int i = st + t; i < st + n; i += THREADS) {
                bool taken = false;
                for (int j = 0; j < r; ++j)
                    if (s_cand[L * TOPK_K + j] == i) { taken = true; break; }
                if (taken) continue;
                const float* an = anchors + (size_t)i * 4;
                float acx = (an[0] + an[2]) * 0.5f;
                float acy = (an[1] + an[3]) * 0.5f;
                float dx = acx - gcx, dy = acy - gcy;
                float d2 = dx * dx + dy * dy;
                unsigned long long key =
                    ((unsigned long long)__float_as_uint(d2) << 32) | (unsigned)i;
                if (key < best) best = key;
            }
            unsigned long long m = blockMinU64(best, s_red);
            if (t == 0) s_cand[L * TOPK_K + r] = (int)(m & 0xffffffffu);
            __syncthreads();
        }
    }

    // candidate IoUs
    if (t < NLEV * TOPK_K) {
        int ai = s_cand[t];
        const float* an = anchors + (size_t)ai * 4;
        float ix1 = fmaxf(an[0], gx1), iy1 = fmaxf(an[1], gy1);
        float ix2 = fminf(an[2], gx2), iy2 = fminf(an[3], gy2);
        float iw = fmaxf(ix2 - ix1, 0.f), ih = fmaxf(iy2 - iy1, 0.f);
        float inter = iw * ih;
        float aa = (an[2] - an[0]) * (an[3] - an[1]);
        float ag = (gx2 - gx1) * (gy2 - gy1);
        float u = fmaxf(aa + ag - inter, EPSF);
        s_iou[t] = inter / u;
    }
    __syncthreads();

    // threshold = mean + std (ddof = 1) over 45 candidates
    if (t == 0) {
        const int NC = NLEV * TOPK_K;
        float s = 0.f;
        for (int i = 0; i < NC; ++i) s += s_iou[i];
        float mean = s / (float)NC;
        float v = 0.f;
        for (int i = 0; i < NC; ++i) { float d = s_iou[i] - mean; v += d * d; }
        v /= (float)(NC - 1);
        s_thr = mean + sqrtf(v);
    }
    __syncthreads();

    // positives: iou >= thr, center strictly inside gt. Scatter argmax key.
    if (t < NLEV * TOPK_K) {
        int ai = s_cand[t];
        const float* an = anchors + (size_t)ai * 4;
        float ccx = (an[0] + an[2]) * 0.5f;
        float ccy = (an[1] + an[3]) * 0.5f;
        float l = ccx - gx1, tt = ccy - gy1, r = gx2 - ccx, bb = gy2 - ccy;
        bool inside = fminf(fminf(l, r), fminf(tt, bb)) > 0.01f;
        if (inside && s_iou[t] >= s_thr) {
            // key = (iou_bits+1)<<32 | (M-1-g): max-key == max iou, ties -> lowest g
            unsigned long long key =
                (((unsigned long long)__float_as_uint(s_iou[t]) + 1ULL) << 32) |
                (unsigned long long)(MGT - 1 - g);
            atomicMax(keys + (size_t)b * A_TOTAL + ai, key);
        }
    }
}

// ---------------- kernel 2: per-anchor reg / ctn / jsd losses + labels ----------------
__global__ void anchor_loss_kernel(const float* __restrict__ pred_reg,
                                   const float* __restrict__ pred_ctn,
                                   const float* __restrict__ anchors,
                                   const float* __restrict__ gt,
                                   const unsigned long long* __restrict__ keys,
                                   int* __restrict__ labels,
                                   float* __restrict__ reg_parts,
                                   float* __restrict__ ctn_parts,
                                   float* __restrict__ jsd_parts,
                                   int*   __restrict__ fg_parts,
                                   int total) {
    __shared__ float s_f[8];
    __shared__ int   s_i[8];
    int idx = blockIdx.x * THREADS + threadIdx.x;
    float lreg = 0.f, lctn = 0.f, ljsd = 0.f;
    int fg = 0, lab = 0;

    if (idx < total) {
        int b = idx / A_TOTAL;
        int a = idx - b * A_TOTAL;
        unsigned long long key = keys[idx];
        if (key != 0ULL) {
            int g = (MGT - 1) - (int)(key & 0xffffffffu);
            const float* G = gt + ((size_t)b * MGT + g) * 5;
            float gx1 = G[0], gy1 = G[1], gx2 = G[2], gy2 = G[3];
            lab = (int)G[4];
            fg  = 1;

            const float* an = anchors + (size_t)a * 4;
            float aw  = an[2] - an[0], ah = an[3] - an[1];
            float acx = (an[0] + an[2]) * 0.5f, acy = (an[1] + an[3]) * 0.5f;

            // encode gt -> bbox target
            float gw = fmaxf(gx2 - gx1, EPSF), gh = fmaxf(gy2 - gy1, EPSF);
            float gcx = (gx1 + gx2) * 0.5f, gcy = (gy1 + gy2) * 0.5f;
            float tx = (gcx - acx) / aw, ty = (gcy - acy) / ah;
            float tw = logf(gw / aw),   th = logf(gh / ah);

            const vf4* pr4 = (const vf4*)pred_reg + (size_t)idx * 2;
            vf4 mu = pr4[0];
            vf4 ls = pr4[1];

            float pbox[4], tbox[4];
            decode_box(mu.x, mu.y, mu.z, mu.w, acx, acy, aw, ah, pbox);
            decode_box(tx,  ty,  tw,  th,  acx, acy, aw, ah, tbox);

            // GIoU regression loss
            lreg = 1.f - giou_fn(pbox, tbox);

            // centerness BCE
            float cl = fmaxf(acx - tbox[0], EPSF);
            float cr = fmaxf(tbox[2] - acx, EPSF);
            float ct = fmaxf(acy - tbox[1], EPSF);
            float cb = fmaxf(tbox[3] - acy, EPSF);
            float ratio = (fminf(cl, cr) / fmaxf(cl, cr)) *
                          (fminf(ct, cb) / fmaxf(ct, cb));
            float ctn = sqrtf(fclampf(ratio, EPSF, 1.f));
            float logit = pred_ctn[idx];
            lctn = fmaxf(logit, 0.f) - logit * ctn + log1pf(expf(-fabsf(logit)));

            // symmetric KL (JSD proxy)
            float tenc[4] = {tx, ty, tw, th};
            float muv[4]  = {mu.x, mu.y, mu.z, mu.w};
            float lsv[4]  = {ls.x, ls.y, ls.z, ls.w};
            float s = 0.f;
#pragma unroll
            for (int j = 0; j < 4; ++j) {
                float var = expf(2.f * lsv[j]);
                float d   = muv[j] - tenc[j];
                float d2  = d * d;
                float kl_pt = -lsv[j] + 0.5f * (var + d2) - 0.5f;
                float kl_tp =  lsv[j] + (1.f + d2) / (2.f * var) - 0.5f;
                s += kl_pt + kl_tp;
            }
            ljsd = 0.5f * s * JS_W;
        }
        labels[idx] = lab;
    }

    lreg = blockReduceSumF(lreg, s_f);
    lctn = blockReduceSumF(lctn, s_f);
    ljsd = blockReduceSumF(ljsd, s_f);
    fg   = blockReduceSumI(fg,   s_i);
    if (threadIdx.x == 0) {
        reg_parts[blockIdx.x] = lreg;
        ctn_parts[blockIdx.x] = lctn;
        jsd_parts[blockIdx.x] = ljsd;
        fg_parts [blockIdx.x] = fg;
    }
}

// ---------------- kernel 3: focal classification loss (bandwidth-bound) ----------------
__global__ void focal_kernel(const float* __restrict__ pred_cls,
                             const int* __restrict__ labels,
                             float* __restrict__ cls_parts,
                             int n4) {
    __shared__ float s_f[8];
    float acc = 0.f;
    int stride = gridDim.x * blockDim.x;
    for (int i4 = blockIdx.x * blockDim.x + threadIdx.x; i4 < n4; i4 += stride) {
        vf4 v = __builtin_nontemporal_load((const vf4*)pred_cls + i4);
        int a   = i4 / (NUM_FG / 4);                    // anchor row
        int c0  = (i4 - a * (NUM_FG / 4)) * 4;          // class of element 0
        int lab = labels[a];
        int tgt = lab - 1;
        float x[4] = {v.x, v.y, v.z, v.w};
#pragma unroll
        for (int j = 0; j < 4; ++j) {
            float p  = 1.f / (1.f + __expf(-x[j]));
            bool  oh = (lab > 0) && (c0 + j == tgt);
            float q  = oh ? p : 1.f - p;                // prob of the "correct" outcome
            float w  = oh ? 0.25f : 0.75f;              // alpha / (1 - alpha)
            float om = 1.f - q;
            acc += -w * om * om * __logf(fmaxf(q, 1e-12f));
        }
    }
    acc = blockReduceSumF(acc, s_f);
    if (threadIdx.x == 0) cls_parts[blockIdx.x] = acc;
}

// ---------------- WMMA-based deterministic wave reduction ----------------
// D = ones(16x4) x B(4x16) + C accumulates the sum of all 64 B-elements into
// every row of C (rows identical), independent of B's exact lane layout.
// n must be a multiple of 64; tail slots of `data` are guaranteed zero (the
// workspace is zero-initialized each call and kernels never write the pad).
__device__ __forceinline__ float wave_sum_f32(const float* __restrict__ data, int n) {
#if defined(__HIP_DEVICE_COMPILE__) && defined(__gfx1250__) && \
    __has_builtin(__builtin_amdgcn_wmma_f32_16x16x4_f32)
    int lane = threadIdx.x & 31;
    v2f av = {1.f, 1.f};
    v8f acc = {};
    const v2f* d2 = (const v2f*)data + lane;   // 8B-aligned: base 16B, lane*8
    for (int base = 0; base < n; base += 64, d2 += 32) {
        v2f bv = *d2;                          // unguarded global_load_b64
        acc = __builtin_amdgcn_wmma_f32_16x16x4_f32(
            /*neg_a=*/false, av, /*neg_b=*/false, bv,
            /*c_mod=*/(short)0, acc, /*reuse_a=*/false, /*reuse_b=*/false);
    }
    // acc[0], lanes 0..15 hold row M=0 (N = lane): sum them -> total
    float v = acc[0];
    v += __shfl_down(v, 8, 16);
    v += __shfl_down(v, 4, 16);
    v += __shfl_down(v, 2, 16);
    v += __shfl_down(v, 1, 16);
    return __shfl(v, 0, 32);
#else
    float s = 0.f;
    if ((threadIdx.x & 31) == 0)
        for (int i = 0; i < n; ++i) s += data[i];
    return __shfl(s, 0, 32);
#endif
}

// ---------------- kernel 4: finalize (single wave; EXEC all ones for WMMA) ----------------
__global__ void finalize_kernel(const float* __restrict__ cls_parts, int n3p,
                                const float* __restrict__ reg_parts,
                                const float* __restrict__ ctn_parts,
                                const float* __restrict__ jsd_parts,
                                const int*   __restrict__ fg_parts, int n2p,
                                float* __restrict__ out) {
    float lcls = wave_sum_f32(cls_parts, n3p);
    float lreg = wave_sum_f32(reg_parts, n2p);
    float lctn = wave_sum_f32(ctn_parts, n2p);
    float ljsd = wave_sum_f32(jsd_parts, n2p);
    int c = 0;
    for (int i = (int)threadIdx.x; i < n2p; i += 32) c += fg_parts[i];  // pad is zero
#pragma unroll
    for (int o = 16; o > 0; o >>= 1) c += __shfl_down(c, o, 32);
    c = __shfl(c, 0, 32);
    if (threadIdx.x == 0) {
        float ln = 0.9f * 100.f + 0.1f * fmaxf((float)c, 1.f);
        out[0] = lcls / ln;
        out[1] = lreg / ln;
        out[2] = lctn / ln;
        out[3] = ljsd / ln;
    }
}

// ---------------- host entry ----------------
extern "C" void kernel_launch(void* const* d_in, const int* in_sizes, int n_in,
                              void* d_out, int out_size, void* d_ws, size_t ws_size,
                              hipStream_t stream) {
    const float* pred_cls = (const float*)d_in[0];
    const float* pred_reg = (const float*)d_in[1];
    const float* pred_ctn = (const float*)d_in[2];
    const float* anchors  = (const float*)d_in[3];
    const float* gt_boxes = (const float*)d_in[4];
    (void)in_sizes; (void)n_in; (void)out_size;

    const int total = BATCH * A_TOTAL;                       // 349184
    const int NB2   = (total + THREADS - 1) / THREADS;       // 1364 blocks
    const int NB2P  = (NB2 + 63) & ~63;                      // 1408, padded to 64

    char* ws = (char*)d_ws;
    size_t off = 0;
    unsigned long long* keys = (unsigned long long*)(ws + off); off += (size_t)total * 8;
    int*   labels    = (int*)  (ws + off); off += (size_t)total * 4;
    float* reg_parts = (float*)(ws + off); off += (size_t)NB2P * 4;
    float* ctn_parts = (float*)(ws + off); off += (size_t)NB2P * 4;
    float* jsd_parts = (float*)(ws + off); off += (size_t)NB2P * 4;
    int*   fg_parts  = (int*)  (ws + off); off += (size_t)NB2P * 4;
    float* cls_parts = (float*)(ws + off); off += (size_t)NB3 * 4;
    (void)ws_size;

    // zero keys + all partial arrays (pads stay zero; kernels overwrite the rest)
    hipMemsetAsync(ws, 0, off, stream);

    assign_kernel<<<BATCH * MGT, THREADS, 0, stream>>>(anchors, gt_boxes, keys);

    anchor_loss_kernel<<<NB2, THREADS, 0, stream>>>(
        pred_reg, pred_ctn, anchors, gt_boxes, keys, labels,
        reg_parts, ctn_parts, jsd_parts, fg_parts, total);

    const int n4 = total * (NUM_FG / 4);                     // float4 elements
    focal_kernel<<<NB3, THREADS, 0, stream>>>(pred_cls, labels, cls_parts, n4);

    finalize_kernel<<<1, 32, 0, stream>>>(cls_parts, NB3,
                                          reg_parts, ctn_parts, jsd_parts,
                                          fg_parts, NB2P, (float*)d_out);
}